// LSTM_72722386256381
// MI455X (gfx1250) — compile-verified
//
#include <hip/hip_runtime.h>
#include <math.h>

// Problem constants (match reference)
#define NB   2048   // batch
#define NT   256    // timesteps
#define NE   128    // embedding dim (= K for input GEMM)
#define NH   64     // hidden
#define NG   256    // 4*H gate width
#define BT   16     // batch rows per workgroup
#define WGSZ 512    // 16 wave32 waves

typedef __attribute__((ext_vector_type(2))) float v2f;
typedef __attribute__((ext_vector_type(4))) float v4f;
typedef __attribute__((ext_vector_type(8))) float v8f;

__device__ __forceinline__ float sigm(float x) { return 1.0f / (1.0f + expf(-x)); }
// numerically stable softplus: log(1+exp(a))
__device__ __forceinline__ float softplus(float a) {
  return fmaxf(a, 0.0f) + log1pf(expf(-fabsf(a)));
}

__global__ void zero_loss_kernel(float* out) { out[NB] = 0.0f; }

__global__ __launch_bounds__(WGSZ)
void lstm_fused_kernel(const int* __restrict__ x, const float* __restrict__ targets,
                       const float* __restrict__ emb, const float* __restrict__ W_ih,
                       const float* __restrict__ W_hh, const float* __restrict__ b_ih,
                       const float* __restrict__ b_hh, const float* __restrict__ W_fc,
                       const float* __restrict__ b_fc, float* __restrict__ out)
{
  __shared__ float xe_lds[2][BT][NE]; // double-buffered embedding tile, 16 KB
  __shared__ float h_lds[BT][NH];     // hidden state, 4 KB
  __shared__ float g_lds[BT][NG];     // pre-activation gates, 16 KB

  const int tid  = threadIdx.x;
  const int wave = tid >> 5;          // 0..15 (wave32)
  const int lane = tid & 31;
  const int b0   = blockIdx.x * BT;

  // WMMA tile coordinates for this wave's 16x16 output tile (N-tile = wave)
  const int nn   = (wave << 4) + (lane & 15);  // gate column owned by this lane
  const int kgrp = (lane >> 4) << 1;           // K sub-offset per f32 A/B layout
  const int am   = lane & 15;                  // A-matrix row for this lane

  // ---- Persistent B panel in registers: [W_ih^T ; W_hh^T] columns nn, K=0..191
  v2f bfrag[48];
#pragma unroll
  for (int kk = 0; kk < 32; ++kk) {            // K = 0..127 from W_ih[nn][k]
    const float* p = W_ih + nn * NE + kk * 4 + kgrp;
    bfrag[kk] = v2f{p[0], p[1]};
  }
#pragma unroll
  for (int kk = 0; kk < 16; ++kk) {            // K = 128..191 from W_hh[nn][k-128]
    const float* p = W_hh + nn * NH + kk * 4 + kgrp;
    bfrag[32 + kk] = v2f{p[0], p[1]};
  }
  const float bias_n = b_ih[nn] + b_hh[nn];

  // ---- init h = 0 in LDS, c = 0 in registers (2 state elems per thread)
  for (int i = tid; i < BT * NH; i += WGSZ) (&h_lds[0][0])[i] = 0.0f;
  float cA = 0.0f, cB = 0.0f;
  const int um = tid >> 6;   // rows um and um+8
  const int uh = tid & 63;   // hidden index

  // ---- prologue: gather t=0 tile (wave == batch row; 512B coalesced row load)
  const int cb = lane << 2;
  {
    const int ix = x[(b0 + wave) * NT + 0];
    *(v4f*)&xe_lds[0][wave][cb] = *(const v4f*)(emb + ix * NE + cb);
  }
  __syncthreads();

  for (int t = 0; t < NT; ++t) {
    const int buf = t & 1;

    // ---- prefetch t+1 embedding row into registers; latency hidden behind
    //      the WMMA chain (clamped redundant load on the final step)
    const int tn = (t + 1 < NT) ? (t + 1) : (NT - 1);
    const int ixn = x[(b0 + wave) * NT + tn];
    const v4f pre = *(const v4f*)(emb + ixn * NE + cb);

    // ---- gates = bias + xe @ W_ih^T + h @ W_hh^T via fp32 WMMA chain
    v8f acc;
#pragma unroll
    for (int j = 0; j < 8; ++j) acc[j] = bias_n;

#pragma unroll
    for (int kk = 0; kk < 32; ++kk) {
      v2f a = *(const v2f*)&xe_lds[buf][am][kk * 4 + kgrp];
      acc = __builtin_amdgcn_wmma_f32_16x16x4_f32(false, a, false, bfrag[kk],
                                                  (short)0, acc, false, false);
    }
#pragma unroll
    for (int kk = 0; kk < 16; ++kk) {
      v2f a = *(const v2f*)&h_lds[am][kk * 4 + kgrp];
      acc = __builtin_amdgcn_wmma_f32_16x16x4_f32(false, a, false, bfrag[32 + kk],
                                                  (short)0, acc, false, false);
    }

    // ---- spill 16x16 tile to LDS: VGPR j -> row (j + 8*(lane>>4)), col nn
#pragma unroll
    for (int j = 0; j < 8; ++j)
      g_lds[j + ((lane >> 4) << 3)][nn] = acc[j];
    __syncthreads();   // gates visible; all xe[buf]/h reads complete

    // ---- LSTM pointwise update: i,f,g,o -> c,h (gate order matches PyTorch)
    {
      float gi = g_lds[um][uh];
      float gf = g_lds[um][NH + uh];
      float gg = g_lds[um][2 * NH + uh];
      float go = g_lds[um][3 * NH + uh];
      cA = sigm(gf) * cA + sigm(gi) * tanhf(gg);
      h_lds[um][uh] = sigm(go) * tanhf(cA);

      gi = g_lds[um + 8][uh];
      gf = g_lds[um + 8][NH + uh];
      gg = g_lds[um + 8][2 * NH + uh];
      go = g_lds[um + 8][3 * NH + uh];
      cB = sigm(gf) * cB + sigm(gi) * tanhf(gg);
      h_lds[um + 8][uh] = sigm(go) * tanhf(cB);
    }

    // ---- commit prefetched tile for t+1 into the alternate buffer
    *(v4f*)&xe_lds[buf ^ 1][wave][cb] = pre;
    __syncthreads();   // h + next xe visible
  }

  // ---- final FC + stable BCE-with-logits (pos_weight=20), mean via atomic
  if (tid < BT) {
    float z = b_fc[0];
#pragma unroll 8
    for (int k = 0; k < NH; ++k) z += h_lds[tid][k] * W_fc[k];
    out[b0 + tid] = z;

    const float tg     = targets[b0 + tid];
    const float ls_pos = -softplus(-z);  // log_sigmoid(z)
    const float ls_neg = -softplus(z);   // log_sigmoid(-z)
    const float term   = -(20.0f * tg * ls_pos + (1.0f - tg) * ls_neg);
    atomicAdd(out + NB, term * (1.0f / (float)NB));
  }
}

extern "C" void kernel_launch(void* const* d_in, const int* in_sizes, int n_in,
                              void* d_out, int out_size, void* d_ws, size_t ws_size,
                              hipStream_t stream) {
  (void)in_sizes; (void)n_in; (void)out_size; (void)d_ws; (void)ws_size;
  const int*   x       = (const int*)d_in[0];
  const float* targets = (const float*)d_in[1];
  const float* emb     = (const float*)d_in[2];
  const float* W_ih    = (const float*)d_in[3];
  const float* W_hh    = (const float*)d_in[4];
  const float* b_ih    = (const float*)d_in[5];
  const float* b_hh    = (const float*)d_in[6];
  const float* W_fc    = (const float*)d_in[7];
  const float* b_fc    = (const float*)d_in[8];
  float* out = (float*)d_out;

  zero_loss_kernel<<<1, 1, 0, stream>>>(out);
  lstm_fused_kernel<<<NB / BT, WGSZ, 0, stream>>>(x, targets, emb, W_ih, W_hh,
                                                  b_ih, b_hh, W_fc, b_fc, out);
}